// SharedEncoder_13675175870684
// MI455X (gfx1250) — compile-verified
//
#include <hip/hip_runtime.h>
#include <hip/hip_bf16.h>

typedef float v2f __attribute__((ext_vector_type(2)));
typedef float v8f __attribute__((ext_vector_type(8)));

#define D 128
#define EPSF 1e-5f

// ---------------- zero ----------------
__global__ void zero_kernel(float* __restrict__ p, size_t n) {
    size_t i = (size_t)blockIdx.x * blockDim.x + threadIdx.x;
    size_t stride = (size_t)gridDim.x * blockDim.x;
    for (; i < n; i += stride) p[i] = 0.0f;
}

// ---------------- edge scatter: agg[dst] += x[src]; deg[dst] += 1 ----------------
__global__ void scatter_kernel(const float* __restrict__ xin,
                               const int* __restrict__ src,
                               const int* __restrict__ dst,
                               float* __restrict__ agg,
                               float* __restrict__ deg,
                               int E, int withDeg) {
    long long gid = (long long)blockIdx.x * blockDim.x + threadIdx.x;
    int e = (int)(gid >> 5);       // 32 threads per edge
    int part = (int)(gid & 31);    // 4 columns per thread
    if (e >= E) return;
    int s = src[e], d = dst[e];
    const float4 v = *(const float4*)(xin + (size_t)s * D + part * 4);
    float* ap = agg + (size_t)d * D + part * 4;
    atomicAdd(ap + 0, v.x);
    atomicAdd(ap + 1, v.y);
    atomicAdd(ap + 2, v.z);
    atomicAdd(ap + 3, v.w);
    if (withDeg && part == 0) atomicAdd(deg + d, 1.0f);
}

// ---------------- WMMA fp32 GEMM over 16-row node stripes ----------------
// out[16 x 128] = (MEAN? A1/max(deg,1) : A1) @ B1  (+ A2 @ B2 if DUAL) + bias
// optional ReLU, optional residual add, optional per-column sum/sumsq atomics (BN stats)
template <bool DUAL, bool MEAN, bool RELU, bool ADDRES, bool STATS>
__global__ __launch_bounds__(128)
void gemm16_kernel(const float* __restrict__ A1, const float* __restrict__ deg,
                   const float* __restrict__ A2,
                   const float* __restrict__ B1, const float* __restrict__ B2,
                   const float* __restrict__ bias, const float* __restrict__ res,
                   float* __restrict__ out,
                   float* __restrict__ colsum, float* __restrict__ colsumsq,
                   int ntiles) {
    const int wave = threadIdx.x >> 5;
    const int tile = blockIdx.x * 4 + wave;     // wave-uniform
    if (tile >= ntiles) return;                  // uniform exit keeps EXEC all-ones
    const int l    = threadIdx.x & 31;
    const int lane16 = l & 15;                   // A-row / B-col / C-col
    const int half = l >> 4;

    const int node = tile * 16 + lane16;
    const size_t rb = (size_t)node * D;

    float dscale = 1.0f;
    if (MEAN) dscale = 1.0f / fmaxf(deg[node], 1.0f);

    v8f acc[8];
    #pragma unroll
    for (int c = 0; c < 8; ++c)
        #pragma unroll
        for (int j = 0; j < 8; ++j) acc[c][j] = 0.0f;

    #pragma unroll 1
    for (int kt = 0; kt < D / 4; ++kt) {
        const int kbase = kt * 4 + half * 2;
        v2f a1;
        a1.x = A1[rb + kbase];
        a1.y = A1[rb + kbase + 1];
        if (MEAN) { a1.x *= dscale; a1.y *= dscale; }
        v2f a2;
        if (DUAL) {
            a2.x = A2[rb + kbase];
            a2.y = A2[rb + kbase + 1];
        }
        #pragma unroll
        for (int ct = 0; ct < 8; ++ct) {
            const int n = ct * 16 + lane16;
            v2f b1v;
            b1v.x = B1[(size_t)kbase * D + n];
            b1v.y = B1[(size_t)(kbase + 1) * D + n];
            acc[ct] = __builtin_amdgcn_wmma_f32_16x16x4_f32(
                false, a1, false, b1v, (short)0, acc[ct], false, false);
            if (DUAL) {
                v2f b2v;
                b2v.x = B2[(size_t)kbase * D + n];
                b2v.y = B2[(size_t)(kbase + 1) * D + n];
                acc[ct] = __builtin_amdgcn_wmma_f32_16x16x4_f32(
                    false, a2, false, b2v, (short)0, acc[ct], false, false);
            }
        }
    }

    // epilogue: bias, activation, residual, store, BN-stat atomics
    #pragma unroll
    for (int ct = 0; ct < 8; ++ct) {
        const int n = ct * 16 + lane16;
        const float bv = bias[n];
        float psum = 0.0f, psq = 0.0f;
        #pragma unroll
        for (int i = 0; i < 8; ++i) {
            const int m = i + half * 8;
            const size_t idx = ((size_t)(tile * 16 + m)) * D + n;
            float v = acc[ct][i] + bv;
            if (RELU) v = fmaxf(v, 0.0f);
            if (ADDRES) v += res[idx];
            out[idx] = v;
            if (STATS) { psum += v; psq += v * v; }
        }
        if (STATS) {
            atomicAdd(colsum + n, psum);
            atomicAdd(colsumsq + n, psq);
        }
    }
}

// ---------------- BN (biased var) + ReLU ----------------
__global__ void bn_relu_kernel(const float* __restrict__ s,
                               const float* __restrict__ gamma,
                               const float* __restrict__ beta,
                               const float* __restrict__ colsum,
                               const float* __restrict__ colsumsq,
                               float* __restrict__ out,
                               size_t total, float invN) {
    size_t i = (size_t)blockIdx.x * blockDim.x + threadIdx.x;
    if (i >= total) return;
    const int c = (int)(i & (D - 1));
    const float mu = colsum[c] * invN;
    const float var = colsumsq[c] * invN - mu * mu;
    const float y = gamma[c] * (s[i] - mu) * rsqrtf(var + EPSF) + beta[c];
    out[i] = fmaxf(y, 0.0f);
}

extern "C" void kernel_launch(void* const* d_in, const int* in_sizes, int n_in,
                              void* d_out, int out_size, void* d_ws, size_t ws_size,
                              hipStream_t stream) {
    const float* x   = (const float*)d_in[0];
    const int* ei    = (const int*)d_in[1];
    const float* W1l = (const float*)d_in[2];
    const float* b1  = (const float*)d_in[3];
    const float* W1r = (const float*)d_in[4];
    const float* g1  = (const float*)d_in[5];
    const float* be1 = (const float*)d_in[6];
    const float* W2l = (const float*)d_in[7];
    const float* b2  = (const float*)d_in[8];
    const float* W2r = (const float*)d_in[9];
    const float* g2  = (const float*)d_in[10];
    const float* be2 = (const float*)d_in[11];
    const float* Wm1 = (const float*)d_in[12];
    const float* bm1 = (const float*)d_in[13];
    const float* Wm2 = (const float*)d_in[14];
    const float* bm2 = (const float*)d_in[15];
    float* outp = (float*)d_out;

    const int N = in_sizes[0] / D;
    const int E = in_sizes[1] / 2;
    const int* src = ei;
    const int* dst = ei + E;

    const size_t nd = (size_t)N * D;
    float* ws   = (float*)d_ws;
    float* agg  = ws;             // reused as MLP hidden later
    float* sbuf = ws + nd;
    float* x1   = ws + 2 * nd;
    float* x2   = ws + 3 * nd;
    float* deg  = ws + 4 * nd;
    float* cs1  = deg + N;
    float* cq1  = cs1 + D;
    float* cs2  = cq1 + D;
    float* cq2  = cs2 + D;

    const int ntiles = (N + 15) / 16;
    const dim3 gGemm((ntiles + 3) / 4), bGemm(128);
    const long long scatterThreads = (long long)E * 32;
    const dim3 gScat((unsigned)((scatterThreads + 255) / 256)), bScat(256);
    const dim3 gBn((unsigned)((nd + 255) / 256)), bBn(256);
    const float invN = 1.0f / (float)N;

    // ---- layer 1 ----
    zero_kernel<<<1024, 256, 0, stream>>>(agg, nd);
    zero_kernel<<<64, 256, 0, stream>>>(deg, (size_t)N + 4 * D);
    scatter_kernel<<<gScat, bScat, 0, stream>>>(x, src, dst, agg, deg, E, 1);
    gemm16_kernel<true, true, false, false, true><<<gGemm, bGemm, 0, stream>>>(
        agg, deg, x, W1l, W1r, b1, nullptr, sbuf, cs1, cq1, ntiles);
    bn_relu_kernel<<<gBn, bBn, 0, stream>>>(sbuf, g1, be1, cs1, cq1, x1, nd, invN);

    // ---- layer 2 ----
    zero_kernel<<<1024, 256, 0, stream>>>(agg, nd);
    scatter_kernel<<<gScat, bScat, 0, stream>>>(x1, src, dst, agg, nullptr, E, 0);
    gemm16_kernel<true, true, false, false, true><<<gGemm, bGemm, 0, stream>>>(
        agg, deg, x1, W2l, W2r, b2, nullptr, sbuf, cs2, cq2, ntiles);
    bn_relu_kernel<<<gBn, bBn, 0, stream>>>(sbuf, g2, be2, cs2, cq2, x2, nd, invN);

    // ---- residual MLP ----
    gemm16_kernel<false, false, true, false, false><<<gGemm, bGemm, 0, stream>>>(
        x2, nullptr, nullptr, Wm1, nullptr, bm1, nullptr, agg /*hidden*/,
        nullptr, nullptr, ntiles);
    gemm16_kernel<false, false, false, true, false><<<gGemm, bGemm, 0, stream>>>(
        agg, nullptr, nullptr, Wm2, nullptr, bm2, x1 /*residual*/, outp,
        nullptr, nullptr, ntiles);
}